// YoloLoss_68831145886137
// MI455X (gfx1250) — compile-verified
//
#include <hip/hip_runtime.h>

typedef __attribute__((ext_vector_type(2))) float v2f;
typedef __attribute__((ext_vector_type(8))) float v8f;

#define NN    16
#define SS    26
#define BBX   3
#define NCLS  80
#define KK    (SS*SS*BBX)      // 2028
#define C5    (5+NCLS)         // 85
#define NKK   (NN*KK)          // 32448
#define IOU_T 0.6f
#define L_COORD 1.0f
#define L_NOOBJ 1.0f
#define L_OBJ   5.0f
#define L_CLASS 1.0f
#define L_PRIOR 0.01f
#define EPOCH_PRIOR 20
#define NBLK_LOSS 16           // ceil(KK/128)

// ---------------- corner-box precompute ----------------
__global__ void bbox_kernel(const float* __restrict__ yhat, const float* __restrict__ y,
                            float4* __restrict__ predBox, float4* __restrict__ gtBox) {
    int id = blockIdx.x * blockDim.x + threadIdx.x;
    if (id >= NKK) return;
    int k = id % KK;
    int cell = k / BBX;
    int sx = cell % SS, sy = cell / SS;
    const float inv = 1.0f / (float)SS;

    const float* ph = yhat + (size_t)id * C5;
    const float* pt = y    + (size_t)id * C5;

    float cx = ((float)sx + ph[0]) * inv, cy = ((float)sy + ph[1]) * inv;
    float hw = ph[2] * 0.5f, hh = ph[3] * 0.5f;
    predBox[id] = make_float4(cx - hw, cy - hh, cx + hw, cy + hh);

    cx = ((float)sx + pt[0]) * inv; cy = ((float)sy + pt[1]) * inv;
    hw = pt[2] * 0.5f; hh = pt[3] * 0.5f;
    gtBox[id] = make_float4(cx - hw, cy - hh, cx + hw, cy + hh);
}

// ---------------- per-pred max IoU over GT -> no_obj mask ----------------
__global__ __launch_bounds__(256)
void maxiou_kernel(const float4* __restrict__ predBox, const float4* __restrict__ gtBox,
                   float* __restrict__ noObj) {
    __shared__ float sx1[256], sy1[256], sx2[256], sy2[256], sar[256];
    int n = blockIdx.y;
    int tid = threadIdx.x;
    int i = blockIdx.x * 256 + tid;
    int ic = (i < KK) ? i : (KK - 1);
    float4 pb = predBox[n * KK + ic];
    float areaA = (pb.z - pb.x) * (pb.w - pb.y);
    float best = 0.0f;

    for (int jt = 0; jt < KK; jt += 256) {
        int j = jt + tid;
        float4 gb = gtBox[n * KK + ((j < KK) ? j : (KK - 1))];
        if (jt + 512 < KK)
            __builtin_prefetch(&gtBox[n * KK + jt + 256 + tid], 0, 3);
        __syncthreads();
        sx1[tid] = gb.x; sy1[tid] = gb.y; sx2[tid] = gb.z; sy2[tid] = gb.w;
        sar[tid] = (gb.z - gb.x) * (gb.w - gb.y);
        __syncthreads();
        int lim = (KK - jt < 256) ? (KK - jt) : 256;
        #pragma unroll 4
        for (int jj = 0; jj < lim; ++jj) {
            float wi = fmaxf(fminf(pb.z, sx2[jj]) - fmaxf(pb.x, sx1[jj]), 0.0f);
            float hi = fmaxf(fminf(pb.w, sy2[jj]) - fmaxf(pb.y, sy1[jj]), 0.0f);
            float inter = wi * hi;
            float iou = inter / (areaA + sar[jj] - inter + 1e-12f);
            best = fmaxf(best, iou);
        }
        __syncthreads();
    }
    if (i < KK) noObj[n * KK + i] = (best < IOU_T) ? 1.0f : 0.0f;
}

// ---------------- per-GT argmax IoU over preds -> idx ----------------
__global__ __launch_bounds__(256)
void argmax_kernel(const float4* __restrict__ predBox, const float4* __restrict__ gtBox,
                   int* __restrict__ idxbuf) {
    __shared__ float sx1[256], sy1[256], sx2[256], sy2[256], sar[256];
    int n = blockIdx.y;
    int tid = threadIdx.x;
    int j = blockIdx.x * 256 + tid;
    int jc = (j < KK) ? j : (KK - 1);
    float4 gb = gtBox[n * KK + jc];
    float areaB = (gb.z - gb.x) * (gb.w - gb.y);
    float best = -1.0f;
    int besti = 0;

    for (int it = 0; it < KK; it += 256) {
        int i = it + tid;
        float4 pb = predBox[n * KK + ((i < KK) ? i : (KK - 1))];
        if (it + 512 < KK)
            __builtin_prefetch(&predBox[n * KK + it + 256 + tid], 0, 3);
        __syncthreads();
        sx1[tid] = pb.x; sy1[tid] = pb.y; sx2[tid] = pb.z; sy2[tid] = pb.w;
        sar[tid] = (pb.z - pb.x) * (pb.w - pb.y);
        __syncthreads();
        int lim = (KK - it < 256) ? (KK - it) : 256;
        #pragma unroll 4
        for (int ii = 0; ii < lim; ++ii) {
            float wi = fmaxf(fminf(gb.z, sx2[ii]) - fmaxf(gb.x, sx1[ii]), 0.0f);
            float hi = fmaxf(fminf(gb.w, sy2[ii]) - fmaxf(gb.y, sy1[ii]), 0.0f);
            float inter = wi * hi;
            float iou = inter / (sar[ii] + areaB - inter + 1e-12f);
            if (iou > best) { best = iou; besti = it + ii; }   // first-max kept
        }
        __syncthreads();
    }
    if (j < KK) idxbuf[n * KK + j] = besti;
}

// ---------------- main loss: scalar terms + WMMA Gram class loss ----------------
__global__ __launch_bounds__(128)
void loss_kernel(const float* __restrict__ yhat, const float* __restrict__ y,
                 const float* __restrict__ anchors, const int* __restrict__ epoch,
                 const int* __restrict__ idxbuf, const float* __restrict__ noObj,
                 float* __restrict__ partials) {
    __shared__ float diffs[128][NCLS + 1];   // +1 pad vs bank conflicts
    __shared__ float gram[4][256];
    __shared__ float red[128];

    int tid = threadIdx.x;
    int n = blockIdx.y;
    int blk = blockIdx.x;
    int k = blk * 128 + tid;
    float vmask = (k < KK) ? 1.0f : 0.0f;
    int kc = (k < KK) ? k : (KK - 1);
    int b = kc % BBX;

    const float* pt = y    + (size_t)(n * KK + kc) * C5;   // GT slot
    const float* po = yhat + (size_t)(n * KK + kc) * C5;   // own prediction
    int g = idxbuf[n * KK + kc];                           // responsible pred index
    const float* pr = yhat + (size_t)(n * KK + g) * C5;    // responsible prediction
    int bg = g % BBX;                                      // anchor of SOURCE box

    float yt0 = pt[0], yt1 = pt[1], yt2 = pt[2], yt3 = pt[3], yt4 = pt[4];
    float have = ((yt4 > 0.0f) ? 1.0f : 0.0f) * vmask;
    float s2 = 2.0f - yt2 * yt3;
    float aw  = anchors[2 * b],  ah  = anchors[2 * b + 1];
    float awg = anchors[2 * bg], ahg = anchors[2 * bg + 1];

    float whtw = logf(yt2 / aw + 1e-16f);
    float whth = logf(yt3 / ah + 1e-16f);
    float r0 = pr[0], r1 = pr[1], r2 = pr[2], r3 = pr[3], r4 = pr[4];
    float whw = logf(r2 / awg + 1e-16f);
    float whh = logf(r3 / ahg + 1e-16f);

    float dx = (r0 - yt0) * s2, dy = (r1 - yt1) * s2;
    float dw = (whw - whtw) * s2, dh = (whh - whth) * s2;
    float coord  = (dx*dx + dy*dy + dw*dw + dh*dh) * have * L_COORD * s2;
    float objl   = (r4 - yt4) * (r4 - yt4) * have * L_OBJ;
    float oc     = po[4] - yt4;
    float noobjl = oc * oc * noObj[n * KK + kc] * L_NOOBJ * vmask;
    float pw = po[2] - aw, ph2 = po[3] - ah;
    float priorOn = (*epoch < EPOCH_PRIOR) ? 1.0f : 0.0f;
    float priorl = (pw * pw + ph2 * ph2) * L_PRIOR * priorOn * vmask;

    // stage class difference rows (zeroed for tail threads -> zero Gram rows)
    #pragma unroll 4
    for (int c = 0; c < NCLS; ++c)
        diffs[tid][c] = (pr[5 + c] - pt[5 + c]) * vmask;
    __syncthreads();

    // Gram diagonal via V_WMMA_F32_16X16X4_F32: for D*D^T the A-operand lane
    // layout (lanes0-15: K0/K1, lanes16-31: K2/K3 of row M) equals the
    // B-operand layout, so one v2f feeds both. 20 chained K=4 steps cover 80.
    int wave = tid >> 5, lane = tid & 31;
    int khi = (lane >> 4) * 2;
    float classSq = 0.0f;
    for (int t = 0; t < 2; ++t) {               // 2 tiles of 16 boxes per wave
        int row = wave * 32 + t * 16 + (lane & 15);
        v8f acc = {};
        #pragma unroll
        for (int s = 0; s < NCLS / 4; ++s) {
            v2f a;
            a.x = diffs[row][4 * s + khi];
            a.y = diffs[row][4 * s + khi + 1];
            acc = __builtin_amdgcn_wmma_f32_16x16x4_f32(
                false, a, false, a, (short)0, acc, false, false);
        }
        __syncthreads();
        #pragma unroll
        for (int r = 0; r < 8; ++r) gram[wave][lane * 8 + r] = acc[r];
        __syncthreads();
        if ((lane >> 4) == t) {                 // owner thread pulls its diagonal
            int m = lane & 15;
            int sl = (m < 8) ? m : (m + 16);    // C/D layout: M>=8 on lanes16-31
            int rr = (m < 8) ? m : (m - 8);
            classSq = gram[wave][sl * 8 + rr];
        }
        __syncthreads();
    }
    float classl = classSq * have * L_CLASS;

    // deterministic block tree-reduction -> per-block partials (no atomics)
    float vals[5] = { coord, classl, noobjl, objl, priorl };
    #pragma unroll
    for (int t = 0; t < 5; ++t) {
        red[tid] = vals[t];
        __syncthreads();
        for (int s = 64; s > 0; s >>= 1) {
            if (tid < s) red[tid] += red[tid + s];
            __syncthreads();
        }
        if (tid == 0) partials[((size_t)t * NN + n) * NBLK_LOSS + blk] = red[0];
        __syncthreads();
    }
}

// ---------------- fixed-order finalize (deterministic) ----------------
__global__ void finalize_kernel(const float* __restrict__ partials, float* __restrict__ out) {
    int t = blockIdx.x * blockDim.x + threadIdx.x;
    if (t >= 5 * NN) return;
    float s = 0.0f;
    for (int i = 0; i < NBLK_LOSS; ++i) s += partials[(size_t)t * NBLK_LOSS + i];
    out[t] = s;   // [coord(16) | class(16) | no_obj(16) | obj(16) | prior(16)]
}

extern "C" void kernel_launch(void* const* d_in, const int* in_sizes, int n_in,
                              void* d_out, int out_size, void* d_ws, size_t ws_size,
                              hipStream_t stream) {
    const float* yhat    = (const float*)d_in[0];
    const float* y       = (const float*)d_in[1];
    const float* anchors = (const float*)d_in[2];
    const int*   epoch   = (const int*)d_in[3];

    float* ws       = (float*)d_ws;
    float4* predBox = (float4*)ws;                         // NKK float4
    float4* gtBox   = (float4*)(ws + (size_t)NKK * 4);     // NKK float4
    float*  noObj   = ws + (size_t)NKK * 8;                // NKK floats
    int*    idxbuf  = (int*)(noObj + NKK);                 // NKK ints
    float*  partials = (float*)(idxbuf + NKK);             // 5*NN*NBLK_LOSS floats
    float*  out = (float*)d_out;

    bbox_kernel<<<(NKK + 255) / 256, 256, 0, stream>>>(yhat, y, predBox, gtBox);

    dim3 gIoU((KK + 255) / 256, NN);
    maxiou_kernel<<<gIoU, 256, 0, stream>>>(predBox, gtBox, noObj);
    argmax_kernel<<<gIoU, 256, 0, stream>>>(predBox, gtBox, idxbuf);

    dim3 gLoss(NBLK_LOSS, NN);
    loss_kernel<<<gLoss, 128, 0, stream>>>(yhat, y, anchors, epoch, idxbuf, noObj, partials);

    finalize_kernel<<<1, 128, 0, stream>>>(partials, out);
}